// NNHMM_11587821764728
// MI455X (gfx1250) — compile-verified
//
#include <hip/hip_runtime.h>
#include <math.h>

// ---------------------------------------------------------------------------
// HMM forward log-likelihood on gfx1250 (MI455X).
//
// result = logsum( (pi .* e0) @ G_1 @ ... @ G_{T-1} ),  G_t = Trans .* e_t (cols)
// Chunked parallel composition in the (+,*) semiring with log-scale carry:
//   K1: 4096 chunks x 1024 transitions -> 8x8 matrix + log-scale each.
//       Emissions via V_WMMA_F32_16X16X4_F32 (D=4 == WMMA K); per-step state
//       max via DPP8 xor-reductions; recurrence via V_PK_FMA_F32 (packed f32);
//       x stream software-pipelined one 16-step tile ahead.
//   K2: 12 pairwise-matmul tree launches: 4096 -> 1 matrix.
//   K3: apply initial vector, logsumexp, write scalar.
// ---------------------------------------------------------------------------

typedef __attribute__((ext_vector_type(2))) float v2f;
typedef __attribute__((ext_vector_type(8))) float v8f;

#define T_LEN   4194304
#define NSTATE  8
#define NCHUNK  4096
#define CLEN    1024          // transitions per chunk
#define NBLK16  (CLEN / 16)   // 16-step tiles per chunk

#define LOG_2PI 1.8378770664093453f

// workspace float offsets
#define TRANS_OFF 0    // 64: row-softmaxed transition probabilities
#define AW_OFF    64   // 32: a[k][d] = mu * inv_std^2
#define BW_OFF    96   // 32: b[k][d] = -0.5 * inv_std^2
#define BIAS_OFF  128  // 8 : per-state emission bias
#define PARAM_FLOATS 256

// e-buffer geometry (per wave): [chunk][step][16 cols], padded chunk stride
#define EBC 264        // 16*16 + 8 pad -> chunks land on distinct LDS banks

// DPP8 selector: arbitrary permute within 8-lane groups (VALU, no DS/waits)
#define DPP8SEL(a,b,c,d,e,f,g,h) \
  ((a)|((b)<<3)|((c)<<6)|((d)<<9)|((e)<<12)|((f)<<15)|((g)<<18)|((h)<<21))
#define DPP8_XOR1 DPP8SEL(1,0,3,2,5,4,7,6)
#define DPP8_XOR2 DPP8SEL(2,3,0,1,6,7,4,5)
#define DPP8_XOR4 DPP8SEL(4,5,6,7,0,1,2,3)

template <unsigned SEL>
__device__ __forceinline__ float dpp8f(float x) {
  return __uint_as_float(
      __builtin_amdgcn_mov_dpp8(__float_as_uint(x), SEL));
}

__device__ __forceinline__ float grpmax8(float x) {  // max over 8-lane group
  x = fmaxf(x, dpp8f<DPP8_XOR1>(x));
  x = fmaxf(x, dpp8f<DPP8_XOR2>(x));
  x = fmaxf(x, dpp8f<DPP8_XOR4>(x));
  return x;
}

__device__ __forceinline__ float bcast_lane(float x, int l) {
  return __uint_as_float(__builtin_amdgcn_readlane(__float_as_uint(x), l));
}

// ---------------------------------------------------------------------------
// K0: parameter prep (softmax of transitions, emission quadratic weights)
// ---------------------------------------------------------------------------
__global__ __launch_bounds__(64) void hmm_prep(const float* __restrict__ utrans,
                                               const float* __restrict__ means,
                                               const float* __restrict__ log_std,
                                               float* __restrict__ ws) {
  const int tid = threadIdx.x;            // 64 threads: (i,j) of Trans
  // --- row softmax of unnormalized_trans (8-lane groups, aligned) ---
  float v = utrans[tid];
  float m = grpmax8(v);
  float e = __expf(v - m);
  float s = e;
  s += dpp8f<DPP8_XOR1>(s);
  s += dpp8f<DPP8_XOR2>(s);
  s += dpp8f<DPP8_XOR4>(s);
  ws[TRANS_OFF + tid] = e / s;

  // --- emission weights: log_em[t,k] = bias_k + x.a_k + (x*x).b_k ---
  if (tid < 32) {                         // (k,d) pairs, 4-lane groups aligned
    float ls  = fminf(fmaxf(log_std[tid], -8.0f), 5.0f);
    float inv = __expf(-ls);
    float w   = inv * inv;
    float mu  = means[tid];
    ws[AW_OFF + tid] = mu * w;
    ws[BW_OFF + tid] = -0.5f * w;
    float bp = -0.5f * LOG_2PI - ls - 0.5f * mu * mu * w;
    bp += dpp8f<DPP8_XOR1>(bp);
    bp += dpp8f<DPP8_XOR2>(bp);
    if ((tid & 3) == 0) ws[BIAS_OFF + (tid >> 2)] = bp;
  }
}

// ---------------------------------------------------------------------------
// K1: per-chunk 8x8 semiring product. One wave = 4 chunks (lane = chunk*8+row).
// ---------------------------------------------------------------------------
__global__ __launch_bounds__(256) void hmm_chunks(const float* __restrict__ x,
                                                  const float* __restrict__ ws,
                                                  float* __restrict__ mats,
                                                  float* __restrict__ scales) {
  __shared__ __align__(16) float ebuf[8][4 * EBC];

  const int tid  = threadIdx.x;
  const int lane = tid & 31;
  const int wv   = tid >> 5;
  const int wave = blockIdx.x * 8 + wv;
  const int myc  = lane >> 3;             // my chunk slot within wave (0..3)
  const int row  = lane & 7;              // my matrix row
  const int chunk = wave * 4 + myc;

  const int j16 = lane & 15;              // WMMA column (state) / A row select
  const int hi  = lane >> 4;              // WMMA K-half select

  // transition matrix as float2 column-pairs (uniform -> scalar regs):
  // trp[k*4+jp] = { Trans[k][2jp], Trans[k][2jp+1] }
  v2f trp[32];
#pragma unroll
  for (int k = 0; k < 8; ++k)
#pragma unroll
    for (int jp = 0; jp < 4; ++jp)
      trp[k * 4 + jp] =
          (v2f){ws[TRANS_OFF + k * 8 + 2 * jp], ws[TRANS_OFF + k * 8 + 2 * jp + 1]};

  // loop-invariant WMMA B fragments (4x16, rows K: vgpr0={0,2}, vgpr1={1,3})
  float b10 = 0.f, b11 = 0.f, b20 = 0.f, b21 = 0.f, biasj = 0.f;
  if (j16 < 8) {
    b10   = ws[AW_OFF + j16 * 4 + (hi ? 2 : 0)];
    b11   = ws[AW_OFF + j16 * 4 + (hi ? 3 : 1)];
    b20   = ws[BW_OFF + j16 * 4 + (hi ? 2 : 0)];
    b21   = ws[BW_OFF + j16 * 4 + (hi ? 3 : 1)];
    biasj = ws[BIAS_OFF + j16];
  }
  const v2f fb1 = {b10, b11};
  const v2f fb2 = {b20, b21};
  const v8f cbias = {biasj, biasj, biasj, biasj, biasj, biasj, biasj, biasj};

  // chunk matrix row = identity row (as 4 packed f32 pairs); scale accumulator
  v2f rv2[4];
#pragma unroll
  for (int jp = 0; jp < 4; ++jp)
    rv2[jp] = (v2f){(2 * jp == row) ? 1.0f : 0.0f,
                    (2 * jp + 1 == row) ? 1.0f : 0.0f};
  float scale = 0.0f;

  const int tbase = 1 + chunk * CLEN;
  const int ebOff = myc * EBC;

  // software pipeline: preload tile 0 of each chunk's x (one b64 per lane)
  float lax[4], lay[4];
#pragma unroll
  for (int cc = 0; cc < 4; ++cc) {
    int tl = 1 + (wave * 4 + cc) * CLEN + j16;
    if (tl > T_LEN - 1) tl = T_LEN - 1;
    const float* xp = x + (unsigned)(tl * 4 + hi * 2);
    lax[cc] = xp[0];
    lay[cc] = xp[1];
  }

  for (int blk = 0; blk < NBLK16; ++blk) {
    // ---- Phase A: emission probs for 16 steps of each of the 4 chunks ----
#pragma unroll
    for (int cc = 0; cc < 4; ++cc) {
      const int t0 = 1 + (wave * 4 + cc) * CLEN + blk * 16;

      const float ax = lax[cc];
      const float ay = lay[cc];
      const v2f fa1 = {ax, ay};
      const v2f fa2 = {ax * ax, ay * ay};

      // D = X @ B1 + (X^2 @ B2 + bias)   -> 16x8 log-emission tile
      v8f d = __builtin_amdgcn_wmma_f32_16x16x4_f32(
          false, fa2, false, fb2, (short)0, cbias, false, false);
      d = __builtin_amdgcn_wmma_f32_16x16x4_f32(
          false, fa1, false, fb1, (short)0, d, false, false);

      // kick off next tile's x load for this chunk (latency hidden by Phase B)
      {
        int tn = t0 + 16 + j16;
        if (tn > T_LEN - 1) tn = T_LEN - 1;
        const float* xp = x + (unsigned)(tn * 4 + hi * 2);
        lax[cc] = xp[0];
        lay[cc] = xp[1];
      }

      // per-step max over states via DPP8, exp, stash in LDS (16-wide, no
      // predication: junk cols 8..15 land in never-read slots)
      float lmv[8];
#pragma unroll
      for (int r = 0; r < 8; ++r) {
        const float lm = grpmax8(d[r]);
        ebuf[wv][cc * EBC + (r + 8 * hi) * 16 + j16] = __expf(d[r] - lm);
        lmv[r] = lm;
      }
      // accumulate shifted log-mass; boundary test hoisted (scalar branch)
      float msum = 0.0f;
      if (t0 + 16 <= T_LEN) {
#pragma unroll
        for (int r = 0; r < 8; ++r) msum += lmv[r];
      } else {
#pragma unroll
        for (int r = 0; r < 8; ++r)
          if (t0 + r + 8 * hi < T_LEN) msum += lmv[r];
      }
      const float tA = bcast_lane(msum, 0);
      const float tB = bcast_lane(msum, 16);
      if (myc == cc) scale += tA + tB;
    }

    // ---- Phase B: 16 recurrence steps, row' = (row @ Trans) .* e ----
    // packed 2xf32 math -> v_pk_fma_f32 (32 packed FMA/step instead of 64)
    const int tt0 = tbase + blk * 16;
    auto stepf = [&](int s) {
      const v2f* ep = (const v2f*)&ebuf[wv][ebOff + s * 16];
      v2f acc[4];
      {
        const float r0 = rv2[0].x;
        const v2f rkk = {r0, r0};
#pragma unroll
        for (int jp = 0; jp < 4; ++jp) acc[jp] = rkk * trp[jp];
      }
#pragma unroll
      for (int k = 1; k < 8; ++k) {
        const float rk = (k & 1) ? rv2[k >> 1].y : rv2[k >> 1].x;
        const v2f rkk = {rk, rk};
#pragma unroll
        for (int jp = 0; jp < 4; ++jp) acc[jp] += rkk * trp[k * 4 + jp];
      }
#pragma unroll
      for (int jp = 0; jp < 4; ++jp) rv2[jp] = acc[jp] * ep[jp];
    };
    if (tt0 + 16 <= T_LEN) {
      for (int s = 0; s < 16; ++s) stepf(s);
    } else {
      for (int s = 0; s < 16; ++s)
        if (tt0 + s < T_LEN) stepf(s);
    }

    // ---- per-tile renormalization (consistent across the chunk's 8 lanes) --
    float rm = fmaxf(fmaxf(fmaxf(rv2[0].x, rv2[0].y), fmaxf(rv2[1].x, rv2[1].y)),
                     fmaxf(fmaxf(rv2[2].x, rv2[2].y), fmaxf(rv2[3].x, rv2[3].y)));
    rm = grpmax8(rm);
    if (rm > 0.0f) {
      const float inv = 1.0f / rm;
      const v2f iv = {inv, inv};
#pragma unroll
      for (int jp = 0; jp < 4; ++jp) rv2[jp] *= iv;
      scale += __logf(rm);
    }
  }

  v2f* mp = (v2f*)(mats + (long)chunk * 64 + row * 8);
#pragma unroll
  for (int jp = 0; jp < 4; ++jp) mp[jp] = rv2[jp];
  if (row == 0) scales[chunk] = scale;
}

// ---------------------------------------------------------------------------
// K2: pairwise semiring combine:  out_p = Q_{2p} @ Q_{2p+1}, renormalized.
// ---------------------------------------------------------------------------
__global__ __launch_bounds__(64) void hmm_combine(const float* __restrict__ mi,
                                                  const float* __restrict__ si,
                                                  float* __restrict__ mo,
                                                  float* __restrict__ so,
                                                  int n) {
  __shared__ float As[64], Bs[64], red[64];
  const int tid = threadIdx.x;
  const int i = tid >> 3, j = tid & 7;
  const int p = blockIdx.x;
  const int a = 2 * p, b = 2 * p + 1;

  const float sa = si[a];
  if (b < n) {
    As[tid] = mi[(long)a * 64 + tid];
    Bs[tid] = mi[(long)b * 64 + tid];
    __syncthreads();
    float acc = 0.0f;
#pragma unroll
    for (int k = 0; k < 8; ++k) acc = fmaf(As[i * 8 + k], Bs[k * 8 + j], acc);
    float sc = sa + si[b];
    red[tid] = acc;
    __syncthreads();
    for (int off = 32; off > 0; off >>= 1) {
      if (tid < off) red[tid] = fmaxf(red[tid], red[tid + off]);
      __syncthreads();
    }
    const float m = red[0];
    if (m > 0.0f) { acc /= m; sc += __logf(m); }
    mo[(long)p * 64 + tid] = acc;
    if (tid == 0) so[p] = sc;
  } else {
    mo[(long)p * 64 + tid] = mi[(long)a * 64 + tid];
    if (tid == 0) so[p] = sa;
  }
}

// ---------------------------------------------------------------------------
// K3: apply initial vector (pi .* e0), logsumexp, emit scalar.
// ---------------------------------------------------------------------------
__global__ void hmm_finalize(const float* __restrict__ x,
                             const float* __restrict__ upi,
                             const float* __restrict__ ws,
                             const float* __restrict__ Q,
                             const float* __restrict__ Qs,
                             float* __restrict__ out) {
  if (threadIdx.x != 0 || blockIdx.x != 0) return;
  float mp = upi[0];
  for (int k = 1; k < 8; ++k) mp = fmaxf(mp, upi[k]);
  float se = 0.0f, lp[8];
  for (int k = 0; k < 8; ++k) { lp[k] = upi[k] - mp; se += __expf(lp[k]); }
  const float lse = __logf(se);

  float al[8];
  for (int k = 0; k < 8; ++k) {
    float em = ws[BIAS_OFF + k];
    for (int d = 0; d < 4; ++d) {
      const float xv = x[d];
      em += xv * ws[AW_OFF + k * 4 + d] + xv * xv * ws[BW_OFF + k * 4 + d];
    }
    al[k] = lp[k] - lse + em;
  }
  float m0 = al[0];
  for (int k = 1; k < 8; ++k) m0 = fmaxf(m0, al[k]);
  float v[8];
  for (int k = 0; k < 8; ++k) v[k] = __expf(al[k] - m0);

  float tot = 0.0f;
  for (int j = 0; j < 8; ++j) {
    float acc = 0.0f;
    for (int i = 0; i < 8; ++i) acc = fmaf(v[i], Q[i * 8 + j], acc);
    tot += acc;
  }
  out[0] = __logf(tot) + m0 + Qs[0];
}

// ---------------------------------------------------------------------------
extern "C" void kernel_launch(void* const* d_in, const int* in_sizes, int n_in,
                              void* d_out, int out_size, void* d_ws, size_t ws_size,
                              hipStream_t stream) {
  (void)in_sizes; (void)n_in; (void)out_size; (void)ws_size;
  const float* x      = (const float*)d_in[0];
  const float* utrans = (const float*)d_in[1];
  const float* upi    = (const float*)d_in[2];
  const float* means  = (const float*)d_in[3];
  const float* logstd = (const float*)d_in[4];
  float* ws = (float*)d_ws;

  float* mats0 = ws + PARAM_FLOATS;
  float* sc0   = mats0 + (long)NCHUNK * 64;
  float* mats1 = sc0 + NCHUNK;
  float* sc1   = mats1 + (long)NCHUNK * 64;

  hmm_prep<<<1, 64, 0, stream>>>(utrans, means, logstd, ws);

  // 4096 chunks / 4 per wave / 8 waves per block = 128 blocks of 256
  hmm_chunks<<<NCHUNK / 32, 256, 0, stream>>>(x, ws, mats0, sc0);

  int n = NCHUNK;
  float *ma = mats0, *sa = sc0, *mb = mats1, *sb = sc1;
  while (n > 1) {
    const int m = (n + 1) / 2;
    hmm_combine<<<m, 64, 0, stream>>>(ma, sa, mb, sb, n);
    float* tm = ma; ma = mb; mb = tm;
    float* ts = sa; sa = sb; sb = ts;
    n = m;
  }

  hmm_finalize<<<1, 64, 0, stream>>>(x, upi, ws, ma, sa, (float*)d_out);
}